// MotorLayer_91053306675754
// MI455X (gfx1250) — compile-verified
//
#include <hip/hip_runtime.h>
#include <stdint.h>

// MI455X (gfx1250) bandwidth-bound split dot-product:
// out[0..1] = W[0..1] . X[:,0] ; out[2..3] = W[2..3] . X[:,1]
// 384 MB single-pass stream -> ~16.5 us roofline at 23.3 TB/s.

typedef float v4f __attribute__((ext_vector_type(4)));
typedef int   v4i __attribute__((ext_vector_type(4)));

#define TW 128                 // indices per wave-tile
#define WAVES_PER_BLOCK 8
#define THREADS (WAVES_PER_BLOCK * 32)
#define STAGE_FLOATS 768       // per buffer: 4*TW weights + 2*TW input

// The async builtin takes pointers to <4 x i32>: global (AS1) src, LDS (AS3) dst.
typedef __attribute__((address_space(1))) v4i GV4;  // global <4 x i32>
typedef __attribute__((address_space(3))) v4i LV4;  // LDS    <4 x i32>

#if __has_builtin(__builtin_amdgcn_global_load_async_to_lds_b128)
  #define HAVE_ASYNC_LDS 1
  #pragma message("CDNA5: using global_load_async_to_lds_b128 streaming path")
#else
  #define HAVE_ASYNC_LDS 0
  #pragma message("CDNA5: async-to-LDS builtin NOT available; direct NT-load path")
#endif

#if __has_builtin(__builtin_amdgcn_s_wait_asynccnt)
  #define WAIT_ASYNCCNT(nimm) __builtin_amdgcn_s_wait_asynccnt(nimm)
  #pragma message("CDNA5: using __builtin_amdgcn_s_wait_asynccnt")
#else
  #define WAIT_ASYNCCNT(nimm) asm volatile("s_wait_asynccnt %0" ::"i"(nimm) : "memory")
  #pragma message("CDNA5: using inline-asm s_wait_asynccnt")
#endif

__global__ __launch_bounds__(THREADS) void motor_main(
    const float* __restrict__ in, const float* __restrict__ w,
    float* __restrict__ partials, long long n, int tiles)
{
    const int wave = threadIdx.x >> 5;
    const int lane = threadIdx.x & 31;
    const long long gwave = (long long)blockIdx.x * WAVES_PER_BLOCK + wave;
    const long long seg = gwave * (long long)tiles * TW;   // this wave's first index

    float a0 = 0.f, a1 = 0.f, a2 = 0.f, a3 = 0.f;

#if HAVE_ASYNC_LDS
    // Double-buffered LDS staging, private per wave (ASYNCcnt is per-wave: no barriers).
    __shared__ __align__(16) float stage[WAVES_PER_BLOCK * 2 * STAGE_FLOATS];
    float* mybuf = &stage[wave * 2 * STAGE_FLOATS];

    auto issue = [&](int t, int b) {
        const long long i0 = seg + (long long)t * TW;
        float* dst = mybuf + b * STAGE_FLOATS;
        const float* wp = w + i0 + 4 * lane;
        #pragma unroll
        for (int r = 0; r < 4; ++r) {
            __builtin_amdgcn_global_load_async_to_lds_b128(
                (GV4*)(wp + (long long)r * n),
                (LV4*)(dst + r * TW + 4 * lane), 0, 0);
        }
        const float* ip = in + 2 * i0 + 4 * lane;
        __builtin_amdgcn_global_load_async_to_lds_b128(
            (GV4*)(ip),      (LV4*)(dst + 4 * TW + 4 * lane), 0, 0);
        __builtin_amdgcn_global_load_async_to_lds_b128(
            (GV4*)(ip + TW), (LV4*)(dst + 5 * TW + 4 * lane), 0, 0);
    };

    auto compute = [&](int b) {
        const float* src = mybuf + b * STAGE_FLOATS;
        v4f xA = *(const v4f*)(src + 4 * TW + 8 * lane);      // {x0[i],x1[i],x0[i+1],x1[i+1]}
        v4f xB = *(const v4f*)(src + 4 * TW + 8 * lane + 4);  // {x0[i+2],...}
        v4f w0 = *(const v4f*)(src + 0 * TW + 4 * lane);
        v4f w1 = *(const v4f*)(src + 1 * TW + 4 * lane);
        v4f w2 = *(const v4f*)(src + 2 * TW + 4 * lane);
        v4f w3 = *(const v4f*)(src + 3 * TW + 4 * lane);
        a0 += w0.x * xA.x + w0.y * xA.z + w0.z * xB.x + w0.w * xB.z;
        a1 += w1.x * xA.x + w1.y * xA.z + w1.z * xB.x + w1.w * xB.z;
        a2 += w2.x * xA.y + w2.y * xA.w + w2.z * xB.y + w2.w * xB.w;
        a3 += w3.x * xA.y + w3.y * xA.w + w3.z * xB.y + w3.w * xB.w;
    };

    issue(0, 0);
    for (int t = 0; t < tiles; ++t) {
        if (t + 1 < tiles) {
            issue(t + 1, (t + 1) & 1);
            WAIT_ASYNCCNT(6);          // previous tile's 6 ops done; next tile's 6 in flight
        } else {
            WAIT_ASYNCCNT(0);
        }
        asm volatile("" ::: "memory"); // make async LDS writes visible to the ds_loads below
        compute(t & 1);
    }
#else
    for (int t = 0; t < tiles; ++t) {
        const long long i0 = seg + (long long)t * TW;
        const float* wp = w + i0 + 4 * lane;
        v4f w0 = __builtin_nontemporal_load((const v4f*)(wp));
        v4f w1 = __builtin_nontemporal_load((const v4f*)(wp + n));
        v4f w2 = __builtin_nontemporal_load((const v4f*)(wp + 2 * n));
        v4f w3 = __builtin_nontemporal_load((const v4f*)(wp + 3 * n));
        const float* ip = in + 2 * i0 + 8 * lane;
        v4f xA = __builtin_nontemporal_load((const v4f*)(ip));
        v4f xB = __builtin_nontemporal_load((const v4f*)(ip + 4));
        a0 += w0.x * xA.x + w0.y * xA.z + w0.z * xB.x + w0.w * xB.z;
        a1 += w1.x * xA.x + w1.y * xA.z + w1.z * xB.x + w1.w * xB.z;
        a2 += w2.x * xA.y + w2.y * xA.w + w2.z * xB.y + w2.w * xB.w;
        a3 += w3.x * xA.y + w3.y * xA.w + w3.z * xB.y + w3.w * xB.w;
    }
#endif

    // Wave32 reduction (fixed order -> deterministic across replays)
    #pragma unroll
    for (int off = 16; off; off >>= 1) {
        a0 += __shfl_xor(a0, off, 32);
        a1 += __shfl_xor(a1, off, 32);
        a2 += __shfl_xor(a2, off, 32);
        a3 += __shfl_xor(a3, off, 32);
    }

    __shared__ float red[WAVES_PER_BLOCK][4];
    if (lane == 0) {
        red[wave][0] = a0; red[wave][1] = a1; red[wave][2] = a2; red[wave][3] = a3;
    }
    __syncthreads();
    if (threadIdx.x == 0) {
        float s0 = 0.f, s1 = 0.f, s2 = 0.f, s3 = 0.f;
        for (int v = 0; v < WAVES_PER_BLOCK; ++v) {
            s0 += red[v][0]; s1 += red[v][1]; s2 += red[v][2]; s3 += red[v][3];
        }
        float* p = partials + (size_t)blockIdx.x * 4;
        p[0] = s0; p[1] = s1; p[2] = s2; p[3] = s3;
    }
}

// One small block: wave j (0..3) reduces partials[:,j] in a fixed order, adds the
// (normally empty) scalar tail, writes out[j].
__global__ __launch_bounds__(128) void motor_reduce(
    const float* __restrict__ partials, int nblocks,
    const float* __restrict__ in, const float* __restrict__ w,
    long long n, long long covered, float* __restrict__ out)
{
    const int j = threadIdx.x >> 5;
    const int lane = threadIdx.x & 31;
    float s = 0.f;
    for (int b = lane; b < nblocks; b += 32) s += partials[(size_t)b * 4 + j];
    #pragma unroll
    for (int off = 16; off; off >>= 1) s += __shfl_xor(s, off, 32);
    if (lane == 0) {
        const int col = j >> 1;
        for (long long i = covered; i < n; ++i)
            s += w[(long long)j * n + i] * in[2 * i + col];
        out[j] = s;
    }
}

extern "C" void kernel_launch(void* const* d_in, const int* in_sizes, int n_in,
                              void* d_out, int out_size, void* d_ws, size_t ws_size,
                              hipStream_t stream) {
    const float* in = (const float*)d_in[0];   // (n, 2) fp32
    const float* w  = (const float*)d_in[1];   // (4, n) fp32
    float* out = (float*)d_out;                // 4 fp32
    const long long n = (long long)in_sizes[0] / 2;

    int blocks = 1024;
    while ((size_t)blocks * 4 * sizeof(float) > ws_size && blocks > 1) blocks >>= 1;

    const long long total_waves = (long long)blocks * WAVES_PER_BLOCK;
    const long long per_wave = n / total_waves;
    const int tiles = (int)(per_wave / TW);
    const long long covered = total_waves * (long long)tiles * TW;
    float* partials = (float*)d_ws;

    int used_blocks = 0;
    if (tiles > 0) {
        motor_main<<<blocks, THREADS, 0, stream>>>(in, w, partials, n, tiles);
        used_blocks = blocks;
    }
    motor_reduce<<<1, 128, 0, stream>>>(partials, used_blocks, in, w, n,
                                        tiles > 0 ? covered : 0, out);
}